// EdgeLoss_50869592655043
// MI455X (gfx1250) — compile-verified
//
#include <hip/hip_runtime.h>
#include <hip/hip_bf16.h>
#include <math.h>

// ---------------------------------------------------------------------------
// Problem constants (from reference):
//   pred_sg_up: (B=8, C=21, H=512, W=512) f32
//   edge_v:     (B=8, 512, 512) i32 in [0,256)
//   adj:        (B=8, 252, 252) f32
//   out:        scalar f32
// ---------------------------------------------------------------------------
#define B_       8
#define C_       21
#define HW_      (512 * 512)
#define NSEG     256
#define A_       252          // NSEG - DROP
#define KPAD     24           // C_ padded to multiple of 4 for WMMA K-steps

typedef __attribute__((ext_vector_type(2))) float v2f;
typedef __attribute__((ext_vector_type(8))) float v8f;

__device__ __forceinline__ void atomic_add_lds(float* p, float v) {
    __hip_atomic_fetch_add(p, v, __ATOMIC_RELAXED, __HIP_MEMORY_SCOPE_WORKGROUP);
}
__device__ __forceinline__ void atomic_add_glb(float* p, float v) {
    __hip_atomic_fetch_add(p, v, __ATOMIC_RELAXED, __HIP_MEMORY_SCOPE_AGENT);
}

// ---------------------------------------------------------------------------
// Kernel 0: zero workspace accumulators
// ---------------------------------------------------------------------------
__global__ void k_zero(float* __restrict__ ws, int n) {
    int i = blockIdx.x * blockDim.x + threadIdx.x;
    if (i < n) ws[i] = 0.0f;
}

// ---------------------------------------------------------------------------
// Kernel 1 (memory-bound hot loop, ~176 MB of pred + 8 MB of edge):
// fused softmax over channels + segment accumulation into per-block LDS bins.
// 4 pixels per thread iteration via float4 -> global_load_b128 (16B/vmem op,
// 4x fewer load instructions than b32-per-channel).
// grid = (32 chunks, B), block = 256 (8 waves).
// LDS: 256*21 sum bins + 256 count bins = 22528 B.
// ---------------------------------------------------------------------------
__global__ void k_softseg(const float* __restrict__ pred,
                          const int* __restrict__ edge,
                          float* __restrict__ sums,     // (B, 256, 21)
                          float* __restrict__ cnts) {   // (B, 256)
    __shared__ float lbin[NSEG * C_];
    __shared__ float lcnt[NSEG];

    const int tid = threadIdx.x;
    const int b   = blockIdx.y;

    for (int i = tid; i < NSEG * C_; i += 256) lbin[i] = 0.0f;
    lcnt[tid] = 0.0f;
    __syncthreads();

    const int chunk = HW_ / gridDim.x;          // 8192 pixels / block
    const int base  = blockIdx.x * chunk;
    const float* pb = pred + (size_t)b * C_ * HW_;
    const int*   eb = edge + (size_t)b * HW_;

    for (int off = 4 * tid; off < chunk; off += 4 * 256) {
        const int p = base + off;               // 16B-aligned, coalesced
        float x[4][C_];
#pragma unroll
        for (int c = 0; c < C_; ++c) {
            const float4 v = *(const float4*)(pb + (size_t)c * HW_ + p);
            x[0][c] = v.x; x[1][c] = v.y; x[2][c] = v.z; x[3][c] = v.w;
        }
        const int4 ev = *(const int4*)(eb + p);
        const int e[4] = {ev.x, ev.y, ev.z, ev.w};
#pragma unroll
        for (int j = 0; j < 4; ++j) {
            float mx = -3.4e38f;
#pragma unroll
            for (int c = 0; c < C_; ++c) mx = fmaxf(mx, x[j][c]);
            float sum = 0.0f;
#pragma unroll
            for (int c = 0; c < C_; ++c) {
                x[j][c] = __expf(x[j][c] - mx);
                sum += x[j][c];
            }
            const float inv = 1.0f / sum;
            float* dst = &lbin[e[j] * C_];
#pragma unroll
            for (int c = 0; c < C_; ++c) atomic_add_lds(&dst[c], x[j][c] * inv);
            atomic_add_lds(&lcnt[e[j]], 1.0f);
        }
    }
    __syncthreads();

    float* gs = sums + (size_t)b * NSEG * C_;
    for (int i = tid; i < NSEG * C_; i += 256) atomic_add_glb(&gs[i], lbin[i]);
    atomic_add_glb(&cnts[b * NSEG + tid], lcnt[tid]);
}

// ---------------------------------------------------------------------------
// Kernel 2: segment mean + channel-wise L2 normalize.
// Writes M as (B, KPAD, 256) K-major, zero-padded in K (21..23) and in
// dropped segments (252..255).  One thread per (b, seg).
// ---------------------------------------------------------------------------
__global__ void k_means(const float* __restrict__ sums,
                        const float* __restrict__ cnts,
                        float* __restrict__ Mbuf) {
    const int idx = blockIdx.x * blockDim.x + threadIdx.x;
    if (idx >= B_ * NSEG) return;
    const int b = idx >> 8;
    const int s = idx & (NSEG - 1);

    const float cnt = fmaxf(cnts[b * NSEG + s], 1.0f);
    float mean[C_];
    float nrm2 = 0.0f;
#pragma unroll
    for (int c = 0; c < C_; ++c) {
        float v = sums[(size_t)b * NSEG * C_ + s * C_ + c] / cnt;
        mean[c] = v;
        nrm2 += v * v;
    }
    const float inv = (s < A_) ? (1.0f / (sqrtf(nrm2) + 1e-6f)) : 0.0f;

    float* M = Mbuf + (size_t)b * KPAD * NSEG;
#pragma unroll
    for (int k = 0; k < C_; ++k) M[k * NSEG + s] = mean[k] * inv;
#pragma unroll
    for (int k = C_; k < KPAD; ++k) M[k * NSEG + s] = 0.0f;
}

// ---------------------------------------------------------------------------
// Kernel 3: Gram tiles via V_WMMA_F32_16X16X4_F32, fused with adj masking and
// per-column reduction.  One wave32 per 16x16 output tile; 6 K-steps (KPAD/4).
// A-frag (16x4, MxK): lane m = lane&15, VGPR0 holds K = k0+2*half,
//                     VGPR1 holds K = k0+2*half+1 (ISA 7.12.2 layout).
// B-frag symmetric (Gram: B = same matrix, other tile).
// C/D layout: lane holds column n = lane&15, rows m = r + 8*half.
// Epilogue: clamped-index unconditional loads + select-masking (no exec
// branching), then shfl_xor(16) combines the two row-halves of each column.
// ---------------------------------------------------------------------------
__global__ void k_gram(const float* __restrict__ Mbuf,
                       const float* __restrict__ adj,
                       float* __restrict__ colGA) {     // (B, 256)
    const int tile = blockIdx.x;        // 0..255
    const int b    = blockIdx.y;
    const int ti   = tile >> 4;         // row-tile (m)
    const int tj   = tile & 15;         // col-tile (n)
    const int lane = threadIdx.x;       // 0..31, EXEC all ones
    const int half = lane >> 4;
    const int l15  = lane & 15;

    const float* M = Mbuf + (size_t)b * KPAD * NSEG;

    v8f acc = {};
#pragma unroll
    for (int k0 = 0; k0 < KPAD; k0 += 4) {
        const int kA = k0 + 2 * half;
        v2f a, bb;
        a.x  = M[(kA    ) * NSEG + ti * 16 + l15];
        a.y  = M[(kA + 1) * NSEG + ti * 16 + l15];
        bb.x = M[(kA    ) * NSEG + tj * 16 + l15];
        bb.y = M[(kA + 1) * NSEG + tj * 16 + l15];
        acc = __builtin_amdgcn_wmma_f32_16x16x4_f32(
            /*neg_a=*/false, a, /*neg_b=*/false, bb,
            /*c_mod=*/(short)0, acc, /*reuse_a=*/false, /*reuse_b=*/false);
    }

    const int n  = tj * 16 + l15;
    const int nc = min(n, A_ - 1);
    const float* ab = adj + (size_t)b * A_ * A_;
    float s = 0.0f;
#pragma unroll
    for (int r = 0; r < 8; ++r) {
        const int m  = ti * 16 + r + 8 * half;
        const float av = ab[(size_t)min(m, A_ - 1) * A_ + nc];  // always in-bounds
        s += (m < A_ && n < A_) ? acc[r] * av : 0.0f;           // v_cndmask, no branch
    }
    s += __shfl_xor(s, 16, 32);         // combine the two row-halves of column n
    if (half == 0 && n < A_) atomic_add_glb(&colGA[b * NSEG + n], s);
}

// ---------------------------------------------------------------------------
// Kernel 4: column sums of adj + final ratio mean.  Single block.
// ---------------------------------------------------------------------------
__global__ void k_final(const float* __restrict__ adj,
                        const float* __restrict__ colGA,
                        float* __restrict__ out) {
    __shared__ float red[256];
    const int tid = threadIdx.x;
    float acc = 0.0f;
    for (int idx = tid; idx < B_ * A_; idx += 256) {
        const int b = idx / A_;
        const int n = idx - b * A_;
        const float* ab = adj + (size_t)b * A_ * A_ + n;
        float s = 0.0f;
        for (int m = 0; m < A_; ++m) s += ab[m * A_];   // coalesced across lanes
        acc += colGA[b * NSEG + n] / (s + 1e-6f);
    }
    red[tid] = acc;
    __syncthreads();
    for (int w = 128; w > 0; w >>= 1) {
        if (tid < w) red[tid] += red[tid + w];
        __syncthreads();
    }
    if (tid == 0) out[0] = red[0] / (float)(B_ * A_);
}

// ---------------------------------------------------------------------------
extern "C" void kernel_launch(void* const* d_in, const int* in_sizes, int n_in,
                              void* d_out, int out_size, void* d_ws, size_t ws_size,
                              hipStream_t stream) {
    const float* pred = (const float*)d_in[0];
    const int*   edge = (const int*)d_in[1];
    const float* adj  = (const float*)d_in[2];
    float* out = (float*)d_out;

    float* ws    = (float*)d_ws;
    float* sums  = ws;                               // B*256*21 = 43008
    float* cnts  = sums + B_ * NSEG * C_;            // B*256    = 2048
    float* Mbuf  = cnts + B_ * NSEG;                 // B*24*256 = 49152
    float* colGA = Mbuf + B_ * KPAD * NSEG;          // B*256    = 2048

    const int nz = B_ * NSEG * C_ + B_ * NSEG + B_ * KPAD * NSEG + B_ * NSEG;
    k_zero<<<(nz + 255) / 256, 256, 0, stream>>>(ws, nz);

    dim3 g1(32, B_);
    k_softseg<<<g1, 256, 0, stream>>>(pred, edge, sums, cnts);

    k_means<<<(B_ * NSEG + 255) / 256, 256, 0, stream>>>(sums, cnts, Mbuf);

    dim3 g3(256, B_);
    k_gram<<<g3, 32, 0, stream>>>(Mbuf, adj, colGA);

    k_final<<<1, 256, 0, stream>>>(adj, colGA, out);
}